// MultiTokenMoE_44057774522959
// MI455X (gfx1250) — compile-verified
//
#include <hip/hip_runtime.h>
#include <hip/hip_bf16.h>
#include <math.h>

// Problem constants (match reference)
#define B_   64
#define D_   1024
#define DH_  4096      // 4*D
#define H_   8
#define V_   32000
#define OFF_ 0
#define W_BCE 1.0f
#define W_LB  0.2f
#define W_ENT 0.01f
#define W_BUDGET 0.05f

// d_out layout (floats): logits | gate_logits | active | loss
#define OUT_LOGITS 0
#define OUT_GATE   (B_ * H_ * V_)            // 16384000
#define OUT_ACTIVE (OUT_GATE + B_ * H_)      // 16384512
#define OUT_LOSS   (OUT_ACTIVE + B_ * H_)    // 16385024

// workspace layout (floats)
#define WS_R    0                            // router hidden r [B, 4D]
#define WS_NLL  (B_ * DH_)                   // 512 per-row nll
#define WS_CNT  (WS_NLL + B_ * H_)           // 512 per-row valid
#define WS_AUX  (WS_CNT + B_ * H_)           // 1 aux loss

typedef float v2f __attribute__((ext_vector_type(2)));
typedef float v8f __attribute__((ext_vector_type(8)));

#define KC 64          // K chunk staged in LDS
#define TILE_N 128     // N tile per block (8 waves x 16)

// ---------------------------------------------------------------------------
// Kernel 1: router MLP layer 1 with exact GELU.
// r[64,4096] = gelu(hidden[64,1024] @ W1^T[1024,4096] + b1)
// WMMA f32 16x16x4. W1 is [4096,1024] row-major -> B[k,n] = W1[n*D + k],
// staged through LDS with coalesced loads (transpose in LDS).
// ---------------------------------------------------------------------------
__global__ __launch_bounds__(256) void router_mlp1_kernel(
    const float* __restrict__ hidden, const float* __restrict__ w1,
    const float* __restrict__ b1, float* __restrict__ rbuf) {
  __shared__ float As[KC][72];     // [k][m], padded
  __shared__ float Bs[KC][144];    // [k][n], padded

  const int n0   = blockIdx.x * TILE_N;      // 32 blocks
  const int tid  = threadIdx.x;
  const int wave = tid >> 5;                 // 0..7 -> N subtile
  const int lane = tid & 31;
  const int lo   = lane & 15;
  const int hi   = lane >> 4;

  v8f acc[4];
  #pragma unroll
  for (int m = 0; m < 4; ++m)
    #pragma unroll
    for (int r = 0; r < 8; ++r) acc[m][r] = 0.0f;

  for (int k0 = 0; k0 < D_; k0 += KC) {
    __syncthreads();
    // A chunk: As[k][m] = hidden[m*D + k0+k] (coalesced along k)
    for (int i = tid; i < 64 * KC; i += 256) {
      int m = i >> 6, k = i & 63;
      As[k][m] = hidden[(size_t)m * D_ + k0 + k];
    }
    // B chunk: Bs[k][n] = W1[(n0+n)*D + k0+k] (coalesced along k)
    for (int i = tid; i < TILE_N * KC; i += 256) {
      int n = i >> 6, k = i & 63;
      Bs[k][n] = w1[(size_t)(n0 + n) * D_ + k0 + k];
    }
    __syncthreads();

    #pragma unroll 4
    for (int kk = 0; kk < KC; kk += 4) {
      v2f bfr;
      bfr.x = Bs[kk + 2 * hi][wave * 16 + lo];
      bfr.y = Bs[kk + 2 * hi + 1][wave * 16 + lo];
      #pragma unroll
      for (int m = 0; m < 4; ++m) {
        v2f afr;
        afr.x = As[kk + 2 * hi][m * 16 + lo];
        afr.y = As[kk + 2 * hi + 1][m * 16 + lo];
        acc[m] = __builtin_amdgcn_wmma_f32_16x16x4_f32(
            false, afr, false, bfr, (short)0, acc[m], false, false);
      }
    }
  }

  const int ng = n0 + wave * 16 + lo;
  const float bias = b1[ng];
  #pragma unroll
  for (int m = 0; m < 4; ++m) {
    #pragma unroll
    for (int r = 0; r < 8; ++r) {
      int mg = m * 16 + r + 8 * hi;
      float x = acc[m][r] + bias;
      float g = 0.5f * x * (1.0f + erff(x * 0.70710678118654752f));  // exact GELU
      rbuf[(size_t)mg * DH_ + ng] = g;
    }
  }
}

// ---------------------------------------------------------------------------
// Kernel 2: gate_logits[64,8] = r @ W2^T + b2. One wave per (b,h).
// ---------------------------------------------------------------------------
__global__ __launch_bounds__(256) void gate_kernel(
    const float* __restrict__ rbuf, const float* __restrict__ w2,
    const float* __restrict__ b2, float* __restrict__ gate_out) {
  const int b = blockIdx.x;
  const int h = threadIdx.x >> 5;
  const int lane = threadIdx.x & 31;
  const float* rb = rbuf + (size_t)b * DH_;
  const float* wh = w2 + (size_t)h * DH_;
  float s = 0.0f;
  for (int k = lane; k < DH_; k += 32) s = fmaf(rb[k], wh[k], s);
  #pragma unroll
  for (int off = 16; off > 0; off >>= 1) s += __shfl_xor(s, off, 32);
  if (lane == 0) gate_out[b * H_ + h] = s + b2[h];
}

// ---------------------------------------------------------------------------
// Kernel 3: top-k -> active mask, BCE + load-balance + entropy + budget aux.
// One block of 64 threads (thread b handles one batch row).
// ---------------------------------------------------------------------------
__global__ __launch_bounds__(64) void router_loss_kernel(
    const float* __restrict__ gate, const int* __restrict__ kmax_p,
    float* __restrict__ active_out, float* __restrict__ aux_out) {
  __shared__ float freq[H_];
  __shared__ float red_bce[B_];
  __shared__ float red_ent[B_];
  const int b = threadIdx.x;
  if (b < H_) freq[b] = 0.0f;
  __syncthreads();

  float g[H_];
  #pragma unroll
  for (int h = 0; h < H_; ++h) g[h] = gate[b * H_ + h];

  // top-k (first occurrence on ties, matching lax.top_k)
  const int kmax = kmax_p[0];
  unsigned mask = 0;
  for (int k = 0; k < kmax && k < H_; ++k) {
    int best = 0; float bv = -INFINITY;
    #pragma unroll
    for (int h = 0; h < H_; ++h)
      if (!((mask >> h) & 1u) && g[h] > bv) { bv = g[h]; best = h; }
    mask |= 1u << best;
  }

  // softmax entropy over H
  float mx = -INFINITY;
  #pragma unroll
  for (int h = 0; h < H_; ++h) mx = fmaxf(mx, g[h]);
  float se = 0.0f;
  #pragma unroll
  for (int h = 0; h < H_; ++h) se += expf(g[h] - mx);
  float ent = 0.0f;
  #pragma unroll
  for (int h = 0; h < H_; ++h) {
    float p = expf(g[h] - mx) / se;
    ent -= p * logf(p + 1e-9f);
  }

  // BCE-with-logits vs active mask; accumulate head frequencies
  float bce = 0.0f;
  #pragma unroll
  for (int h = 0; h < H_; ++h) {
    float t = (float)((mask >> h) & 1u);
    active_out[b * H_ + h] = t;
    atomicAdd(&freq[h], t);
    bce += fmaxf(g[h], 0.0f) - g[h] * t + log1pf(expf(-fabsf(g[h])));
  }

  red_bce[b] = bce;
  red_ent[b] = ent;
  __syncthreads();
  for (int s = 32; s > 0; s >>= 1) {
    if (b < s) { red_bce[b] += red_bce[b + s]; red_ent[b] += red_ent[b + s]; }
    __syncthreads();
  }
  if (b == 0) {
    float bcem = red_bce[0] / (float)(B_ * H_);
    float entm = red_ent[0] / (float)B_;
    float lb = 0.0f, act = 0.0f;
    #pragma unroll
    for (int h = 0; h < H_; ++h) {
      float f = freq[h] / (float)B_;
      float d = f - 1.0f / (float)H_;
      lb += d * d;
      act += f;
    }
    float budget = act / (float)H_;
    aux_out[0] = W_BCE * bcem + W_LB * lb + W_ENT * entm + W_BUDGET * budget;
  }
}

// ---------------------------------------------------------------------------
// Kernel 4: the big GEMM. logits[b,h,v] = sum_d hidden[b,d]*heads_w[h,d,v].
// Per block: one head, one 128-wide N tile, full M=64 -> each heads_w element
// read exactly once from HBM (1.05 GB total stream). WMMA f32 16x16x4.
// A staged K-major in LDS; B streamed from global (coalesced); next-chunk
// prefetch via global_prefetch.
// ---------------------------------------------------------------------------
__global__ __launch_bounds__(256) void head_gemm_kernel(
    const float* __restrict__ hidden, const float* __restrict__ heads_w,
    float* __restrict__ logits) {
  __shared__ float As[KC][72];               // [k][m], padded

  const int ntiles = V_ / TILE_N;            // 250
  const int h  = blockIdx.x / ntiles;
  const int n0 = (blockIdx.x % ntiles) * TILE_N;
  const float* Wh = heads_w + (size_t)h * D_ * V_;

  const int tid  = threadIdx.x;
  const int wave = tid >> 5;                 // N subtile 0..7
  const int lane = tid & 31;
  const int lo   = lane & 15;
  const int hi   = lane >> 4;

  v8f acc[4];
  #pragma unroll
  for (int m = 0; m < 4; ++m)
    #pragma unroll
    for (int r = 0; r < 8; ++r) acc[m][r] = 0.0f;

  for (int k0 = 0; k0 < D_; k0 += KC) {
    __syncthreads();
    for (int i = tid; i < 64 * KC; i += 256) {
      int m = i >> 6, k = i & 63;
      As[k][m] = hidden[(size_t)m * D_ + k0 + k];
    }
    __syncthreads();

    // Prefetch next B K-chunk (64 rows x 128 cols = 512 cachelines).
    if (k0 + KC < D_) {
      #pragma unroll
      for (int p = 0; p < 2; ++p) {
        int idx = tid * 2 + p;               // 0..511
        int kr  = idx >> 3;
        int co  = (idx & 7) * 16;
        __builtin_prefetch(Wh + (size_t)(k0 + KC + kr) * V_ + n0 + co, 0, 1);
      }
    }

    const float* Bk = Wh + (size_t)k0 * V_ + n0 + wave * 16;
    #pragma unroll 4
    for (int kk = 0; kk < KC; kk += 4) {
      const float* Bp = Bk + (size_t)kk * V_;
      v2f bfr;                               // B 4x16: VGPR0=K0/K2, VGPR1=K1/K3
      bfr.x = Bp[(size_t)(2 * hi) * V_ + lo];
      bfr.y = Bp[(size_t)(2 * hi + 1) * V_ + lo];
      #pragma unroll
      for (int m = 0; m < 4; ++m) {
        v2f afr;                             // A 16x4: VGPR0=K0/K2, VGPR1=K1/K3
        afr.x = As[kk + 2 * hi][m * 16 + lo];
        afr.y = As[kk + 2 * hi + 1][m * 16 + lo];
        acc[m] = __builtin_amdgcn_wmma_f32_16x16x4_f32(
            false, afr, false, bfr, (short)0, acc[m], false, false);
      }
    }
  }

  const int ng = n0 + wave * 16 + lo;
  #pragma unroll
  for (int m = 0; m < 4; ++m) {
    #pragma unroll
    for (int r = 0; r < 8; ++r) {
      int mg = m * 16 + r + 8 * hi;          // C/D: VGPR r = rows r / r+8
      logits[((size_t)mg * H_ + h) * V_ + ng] = acc[m][r];
    }
  }
}

// ---------------------------------------------------------------------------
// Kernel 5: per (b,h) row: logsumexp over V, masked NLL. One block per row.
// ---------------------------------------------------------------------------
__global__ __launch_bounds__(256) void nll_kernel(
    const float* __restrict__ logits, const int* __restrict__ labels,
    const float* __restrict__ active, float* __restrict__ nll_out,
    float* __restrict__ cnt_out) {
  __shared__ float red[256];
  const int row = blockIdx.x;                // b*H + h
  const int t = threadIdx.x;
  const float* lrow = logits + (size_t)row * V_;

  float m = -INFINITY;
  for (int v = t; v < V_; v += 256) m = fmaxf(m, lrow[v]);
  red[t] = m;
  __syncthreads();
  for (int s = 128; s > 0; s >>= 1) {
    if (t < s) red[t] = fmaxf(red[t], red[t + s]);
    __syncthreads();
  }
  m = red[0];
  __syncthreads();

  float se = 0.0f;
  for (int v = t; v < V_; v += 256) se += expf(lrow[v] - m);
  red[t] = se;
  __syncthreads();
  for (int s = 128; s > 0; s >>= 1) {
    if (t < s) red[t] += red[t + s];
    __syncthreads();
  }

  if (t == 0) {
    int lab = labels[row];
    float nll = -(lrow[lab] - m - logf(red[0]));
    int valid = (active[row] > 0.5f) && (lab != OFF_);
    nll_out[row] = valid ? nll : 0.0f;
    cnt_out[row] = valid ? 1.0f : 0.0f;
  }
}

// ---------------------------------------------------------------------------
// Kernel 6: loss = sum(nll)/max(n_valid,1) + aux
// ---------------------------------------------------------------------------
__global__ __launch_bounds__(512) void final_kernel(
    const float* __restrict__ nll, const float* __restrict__ cnt,
    const float* __restrict__ aux, float* __restrict__ loss_out) {
  __shared__ float r1[512];
  __shared__ float r2[512];
  const int t = threadIdx.x;
  r1[t] = nll[t];
  r2[t] = cnt[t];
  __syncthreads();
  for (int s = 256; s > 0; s >>= 1) {
    if (t < s) { r1[t] += r1[t + s]; r2[t] += r2[t + s]; }
    __syncthreads();
  }
  if (t == 0) loss_out[0] = r1[0] / fmaxf(r2[0], 1.0f) + aux[0];
}

// ---------------------------------------------------------------------------
extern "C" void kernel_launch(void* const* d_in, const int* in_sizes, int n_in,
                              void* d_out, int out_size, void* d_ws, size_t ws_size,
                              hipStream_t stream) {
  const float* hidden   = (const float*)d_in[0];
  const float* w1       = (const float*)d_in[1];
  const float* b1       = (const float*)d_in[2];
  const float* w2       = (const float*)d_in[3];
  const float* b2       = (const float*)d_in[4];
  const float* heads_w  = (const float*)d_in[5];
  const int*   labels   = (const int*)d_in[6];
  const int*   kmax_p   = (const int*)d_in[7];

  float* out = (float*)d_out;
  float* ws  = (float*)d_ws;

  float* logits = out + OUT_LOGITS;
  float* gate   = out + OUT_GATE;
  float* active = out + OUT_ACTIVE;
  float* loss   = out + OUT_LOSS;

  float* rbuf   = ws + WS_R;
  float* nllbuf = ws + WS_NLL;
  float* cntbuf = ws + WS_CNT;
  float* auxbuf = ws + WS_AUX;

  // Router path
  router_mlp1_kernel<<<DH_ / TILE_N, 256, 0, stream>>>(hidden, w1, b1, rbuf);
  gate_kernel<<<B_, 256, 0, stream>>>(rbuf, w2, b2, gate);
  router_loss_kernel<<<1, 64, 0, stream>>>(gate, kmax_p, active, auxbuf);

  // Main head GEMM (HBM-bound: 1.05 GB heads_w stream)
  head_gemm_kernel<<<H_ * (V_ / TILE_N), 256, 0, stream>>>(hidden, heads_w, logits);

  // Masked CE over logits (resident in L2)
  nll_kernel<<<B_ * H_, 256, 0, stream>>>(logits, labels, active, nllbuf, cntbuf);
  final_kernel<<<1, 512, 0, stream>>>(nllbuf, cntbuf, auxbuf, loss);
}